// GCN_197568496077
// MI455X (gfx1250) — compile-verified
//
#include <hip/hip_runtime.h>
#include <hip/hip_bf16.h>

// ---------------------------------------------------------------------------
// 2-layer GCN (PyG GCNConv, normalize=False) for MI455X (gfx1250, wave32).
//
//   h0  = x @ W1                       (WMMA f32 16x16x4, K=128)
//   agg = b1 + scatter_add(h0[src] -> dst)        (L2 fp32 atomics)
//   h2  = relu(agg) @ W2               (WMMA f32 16x16x4, K=64, relu fused)
//   out = b2 + scatter_add(h2[src] -> dst)        (L2 fp32 atomics)
//
// Bottleneck: ~1.3GB irregular edge traffic; scatter targets (<70MB) live in
// the 192MB L2 so atomics run at L2 rate. GEMMs (2.1 GFLOP) use fp32 WMMA for
// reference-precision; A rows are CLAMPED (not predicated) so the inner loop
// has zero branches, W is staged in LDS pre-paired as float2 (ds_load_b64
// into even VGPR pairs), and full tiles take an unguarded store fast path.
// ---------------------------------------------------------------------------

#define IN_DIM  128
#define HID_DIM 64
#define OUT_DIM 40
#define PAD_OUT 48   // OUT_DIM padded to 3 N-tiles of 16

typedef float v2f __attribute__((ext_vector_type(2)));
typedef float v8f __attribute__((ext_vector_type(8)));

// ---------------------------------------------------------------------------
// GEMM1: h0[N,64] = x[N,128] @ W1[128,64]
// 256 threads = 8 waves; each wave owns a 16-row strip and all 4 N-tiles.
// A frag (16x4 f32): lanes 0-15 -> K=k0,k0+1 (M=lid); lanes 16-31 -> K=k0+2,3.
// B frag mirrors with N=lid. LDS holds W1 as float2 K-pairs: sW2[k0/2][n].
// ---------------------------------------------------------------------------
__global__ __launch_bounds__(256) void gcn_gemm1_wmma(
    const float* __restrict__ x, const float* __restrict__ W1,
    float* __restrict__ h0, int N) {
  __shared__ v2f sW2[(IN_DIM / 2) * HID_DIM];              // 32 KB
  for (int i = threadIdx.x; i < (IN_DIM / 2) * HID_DIM; i += 256) {
    const int p = i / HID_DIM;             // K-pair index: K = 2p, 2p+1
    const int n = i % HID_DIM;
    v2f w;
    w.x = W1[(2 * p)     * HID_DIM + n];
    w.y = W1[(2 * p + 1) * HID_DIM + n];
    sW2[i] = w;
  }
  __syncthreads();

  const int wave = threadIdx.x >> 5;
  const int lane = threadIdx.x & 31;
  const int half = lane >> 4;          // 0: K=k0,k0+1  1: K=k0+2,k0+3
  const int lid  = lane & 15;
  const int m0   = blockIdx.x * 128 + wave * 16;

  // Clamp OOB rows to a valid row; their results are discarded at store time.
  const int row = min(m0 + lid, N - 1);
  const float* __restrict__ xrow = x + (size_t)row * IN_DIM;

  v8f acc[4] = {v8f{}, v8f{}, v8f{}, v8f{}};

  for (int kk = 0; kk < IN_DIM; kk += 4) {
    const int k0 = kk + half * 2;
    const v2f a = *(const v2f*)(xrow + k0);          // global_load_b64
    const v2f* __restrict__ wp = sW2 + (size_t)(k0 >> 1) * HID_DIM + lid;
#pragma unroll
    for (int t = 0; t < 4; ++t) {
      const v2f b = wp[t * 16];                      // ds_load_b64
      acc[t] = __builtin_amdgcn_wmma_f32_16x16x4_f32(
          /*neg_a=*/false, a, /*neg_b=*/false, b,
          /*c_mod=*/(short)0, acc[t], /*reuse_a=*/false, /*reuse_b=*/false);
    }
  }

  // C/D layout: VGPR r, lanes 0-15 -> M=r, lanes 16-31 -> M=r+8; N=lid.
  float* __restrict__ hb = h0 + (size_t)(m0 + half * 8) * HID_DIM + lid;
  if (m0 + 16 <= N) {                    // full tile: unguarded, imm offsets
#pragma unroll
    for (int t = 0; t < 4; ++t)
#pragma unroll
      for (int r = 0; r < 8; ++r)
        hb[(size_t)r * HID_DIM + t * 16] = acc[t][r];
  } else {                               // tail tile only (last block)
#pragma unroll
    for (int t = 0; t < 4; ++t)
#pragma unroll
      for (int r = 0; r < 8; ++r) {
        const int rr = m0 + r + half * 8;
        if (rr < N) h0[(size_t)rr * HID_DIM + t * 16 + lid] = acc[t][r];
      }
  }
}

// ---------------------------------------------------------------------------
// GEMM2: h2[N,40] = relu(agg[N,64]) @ W2[64,40]   (N-dim padded 40 -> 48)
// ---------------------------------------------------------------------------
__global__ __launch_bounds__(256) void gcn_gemm2_wmma(
    const float* __restrict__ agg, const float* __restrict__ W2,
    float* __restrict__ h2, int N) {
  __shared__ v2f sW2[(HID_DIM / 2) * PAD_OUT];             // 12 KB, zero-padded
  for (int i = threadIdx.x; i < (HID_DIM / 2) * PAD_OUT; i += 256) {
    const int p = i / PAD_OUT;             // K-pair index
    const int n = i % PAD_OUT;
    v2f w = {0.0f, 0.0f};
    if (n < OUT_DIM) {
      w.x = W2[(2 * p)     * OUT_DIM + n];
      w.y = W2[(2 * p + 1) * OUT_DIM + n];
    }
    sW2[i] = w;
  }
  __syncthreads();

  const int wave = threadIdx.x >> 5;
  const int lane = threadIdx.x & 31;
  const int half = lane >> 4;
  const int lid  = lane & 15;
  const int m0   = blockIdx.x * 128 + wave * 16;

  const int row = min(m0 + lid, N - 1);
  const float* __restrict__ arow = agg + (size_t)row * HID_DIM;

  v8f acc[3] = {v8f{}, v8f{}, v8f{}};

  for (int kk = 0; kk < HID_DIM; kk += 4) {
    const int k0 = kk + half * 2;
    v2f a = *(const v2f*)(arow + k0);                // global_load_b64
    a.x = fmaxf(a.x, 0.0f);                          // ReLU fused
    a.y = fmaxf(a.y, 0.0f);
    const v2f* __restrict__ wp = sW2 + (size_t)(k0 >> 1) * PAD_OUT + lid;
#pragma unroll
    for (int t = 0; t < 3; ++t) {
      const v2f b = wp[t * 16];                      // ds_load_b64
      acc[t] = __builtin_amdgcn_wmma_f32_16x16x4_f32(
          false, a, false, b, (short)0, acc[t], false, false);
    }
  }

  float* __restrict__ hb = h2 + (size_t)(m0 + half * 8) * OUT_DIM + lid;
  if (m0 + 16 <= N) {                    // full tile: only col guard remains
#pragma unroll
    for (int t = 0; t < 3; ++t) {
#pragma unroll
      for (int r = 0; r < 8; ++r) {
        if (t * 16 + lid < OUT_DIM)      // folds: t<2 always true, t==2 lid<8
          hb[(size_t)r * OUT_DIM + t * 16] = acc[t][r];
      }
    }
  } else {
#pragma unroll
    for (int t = 0; t < 3; ++t)
#pragma unroll
      for (int r = 0; r < 8; ++r) {
        const int rr  = m0 + r + half * 8;
        const int col = t * 16 + lid;
        if (rr < N && col < OUT_DIM) h2[(size_t)rr * OUT_DIM + col] = acc[t][r];
      }
  }
}

// ---------------------------------------------------------------------------
// Buffer init with broadcast bias: dst[i] = b[i % D]  (fuses "+ b" pass)
// ---------------------------------------------------------------------------
__global__ __launch_bounds__(256) void gcn_init_bias(
    float* __restrict__ dst, const float* __restrict__ b, int D, long total) {
  const long i = (long)blockIdx.x * 256 + threadIdx.x;
  if (i < total) dst[i] = b[(int)(i % D)];
}

// ---------------------------------------------------------------------------
// Scatter layer 1: one WAVE per edge -> edge indices are wave-uniform, so
// src/dst lower to scalar loads and atomics use SGPR base + lane offset.
// 32 lanes x float2 per edge (coalesced 256B row gathers).
// ---------------------------------------------------------------------------
__global__ __launch_bounds__(256) void gcn_scatter64(
    const float* __restrict__ h0, const int* __restrict__ src,
    const int* __restrict__ dst, float* __restrict__ agg, int E) {
  const int e = blockIdx.x * 8 + (threadIdx.x >> 5);   // wave-uniform edge id
  const int c = (threadIdx.x & 31) * 2;
  if (e < E) {
    const int s = src[e], d = dst[e];                  // scalar (uniform) loads
    const float2 v = *(const float2*)(h0 + (size_t)s * HID_DIM + c);
    float* a = agg + (size_t)d * HID_DIM + c;
    unsafeAtomicAdd(a,     v.x);                       // global_atomic_add_f32
    unsafeAtomicAdd(a + 1, v.y);
  }
}

// ---------------------------------------------------------------------------
// Scatter layer 2: 8 lanes per edge, 5 floats per lane (40-wide rows).
// ---------------------------------------------------------------------------
__global__ __launch_bounds__(256) void gcn_scatter40(
    const float* __restrict__ h2, const int* __restrict__ src,
    const int* __restrict__ dst, float* __restrict__ out, int E) {
  const long t = (long)blockIdx.x * 256 + threadIdx.x;
  const long e = t >> 3;
  const int  c = ((int)t & 7) * 5;
  if (e < E) {
    const int s = src[e], d = dst[e];
    const float* h = h2  + (size_t)s * OUT_DIM + c;
    float*       o = out + (size_t)d * OUT_DIM + c;
#pragma unroll
    for (int j = 0; j < 5; ++j) unsafeAtomicAdd(o + j, h[j]);
  }
}

// ---------------------------------------------------------------------------
extern "C" void kernel_launch(void* const* d_in, const int* in_sizes, int n_in,
                              void* d_out, int out_size, void* d_ws, size_t ws_size,
                              hipStream_t stream) {
  const float* x  = (const float*)d_in[0];   // [N,128]
  const float* W1 = (const float*)d_in[1];   // [128,64]
  const float* b1 = (const float*)d_in[2];   // [64]
  const float* W2 = (const float*)d_in[3];   // [64,40]
  const float* b2 = (const float*)d_in[4];   // [40]
  const int*   ei = (const int*)d_in[5];     // [2,E] (jax default int32)

  const int N = in_sizes[0] / IN_DIM;
  const int E = in_sizes[5] / 2;
  const int* src = ei;
  const int* dst = ei + E;

  float* ws   = (float*)d_ws;
  float* h0   = ws;                          // [N,64]
  float* agg1 = ws + (size_t)N * HID_DIM;    // [N,64]
  float* h2   = agg1 + (size_t)N * HID_DIM;  // [N,40]
  float* out  = (float*)d_out;               // [N,40]

  const int gemmGrid = (N + 127) / 128;

  // h0 = x @ W1
  gcn_gemm1_wmma<<<gemmGrid, 256, 0, stream>>>(x, W1, h0, N);

  // agg1 = b1 (broadcast), then += scatter of h0 over edges
  {
    const long total = (long)N * HID_DIM;
    gcn_init_bias<<<(int)((total + 255) / 256), 256, 0, stream>>>(agg1, b1, HID_DIM, total);
    gcn_scatter64<<<(E + 7) / 8, 256, 0, stream>>>(h0, src, dst, agg1, E);
  }

  // h2 = relu(agg1) @ W2
  gcn_gemm2_wmma<<<gemmGrid, 256, 0, stream>>>(agg1, W2, h2, N);

  // out = b2 (broadcast), then += scatter of h2 over edges
  {
    const long total = (long)N * OUT_DIM;
    gcn_init_bias<<<(int)((total + 255) / 256), 256, 0, stream>>>(out, b2, OUT_DIM, total);
    const long thr = (long)E * 8;
    gcn_scatter40<<<(int)((thr + 255) / 256), 256, 0, stream>>>(h2, src, dst, out, E);
  }
}